// DenoisingLossDDP_1864015806519
// MI455X (gfx1250) — compile-verified
//
#include <hip/hip_runtime.h>

typedef __attribute__((ext_vector_type(2))) float v2f;
typedef __attribute__((ext_vector_type(8))) float v8f;

#define N_T   16
#define BSZ   128
#define DDIM  4096
#define KTOT  (N_T * BSZ)          // 2048 rows
#define PIX   12288                // 3*64*64
#define TILE  128
#define KC    32
#define LDSW  36                   // stride 36 floats: 16B-aligned float4 stores,
                                   // 8B-aligned b64 frag reads, 16 distinct banks

// ---------------------------------------------------------------- init
__global__ void init_out_kernel(float* out) {
    if (threadIdx.x == 0 && blockIdx.x == 0) out[0] = 0.0f;
}

// ------------------------------------------------- per-row scale factor
// scale[i] = (1 / max(||h_i||, eps)) * sqrt(1/TEMP)   (TEMP = 0.5)
__global__ __launch_bounds__(256) void row_scale_kernel(
        const float* __restrict__ h, float* __restrict__ scale) {
    __shared__ float red[256];
    const int row = blockIdx.x;
    const float* r = h + (size_t)row * DDIM;
    float ss = 0.0f;
    for (int c = threadIdx.x; c < DDIM; c += 256) {
        float v = r[c];
        ss += v * v;
    }
    red[threadIdx.x] = ss;
    __syncthreads();
    for (int s = 128; s > 0; s >>= 1) {
        if (threadIdx.x < s) red[threadIdx.x] += red[threadIdx.x + s];
        __syncthreads();
    }
    if (threadIdx.x == 0) {
        float n = fmaxf(sqrtf(red[0]), 1e-8f);
        scale[row] = (1.0f / n) * 1.41421356237309504880f;  // * sqrt(1/0.5)
    }
}

// ---------------------------------------------------- Gram via f32 WMMA
// simmat = (sn sn^T)/TEMP. Raw h tiles staged in LDS (pure copy), rank-1
// normalization applied in the epilogue: out = acc * s_row * s_col.
// Block: 128x128 tile, 8 waves in a 4x2 grid, per-wave tile 32x64
// (8 v8f accumulators; per k-step: 2 A-frags + 4 B-frags -> 8 WMMAs).
__global__ __launch_bounds__(256) void gram_wmma_kernel(
        const float* __restrict__ h, const float* __restrict__ scale,
        float* __restrict__ simmat) {
    __shared__ float As[TILE][LDSW];
    __shared__ float Bs[TILE][LDSW];

    const int row0 = blockIdx.y * TILE;
    const int col0 = blockIdx.x * TILE;
    const int tid  = threadIdx.x;
    const int lane = tid & 31;
    const int wave = tid >> 5;        // 0..7
    const int half = lane >> 4;       // 0/1
    const int m16  = lane & 15;

    const int wrow = wave >> 1;       // 0..3
    const int wcol = wave & 1;        // 0..1
    const int r0w  = wrow * 32;       // wave row base in tile
    const int c0w  = wcol * 64;       // wave col base in tile

    // staging: 256 threads cover 128 rows x 32 cols (one float4 each, x4 rows)
    const int lr = tid >> 3;          // 0..31
    const int lk = (tid & 7) * 4;     // 0..28

    v8f acc[8];
#pragma unroll
    for (int t = 0; t < 8; ++t) {
        v8f z = {0.f, 0.f, 0.f, 0.f, 0.f, 0.f, 0.f, 0.f};
        acc[t] = z;
    }

    for (int kb = 0; kb < DDIM; kb += KC) {
        __syncthreads();
#pragma unroll
        for (int q = 0; q < 4; ++q) {
            const int r = lr + 32 * q;
            const float4 va = *(const float4*)&h[(size_t)(row0 + r) * DDIM + kb + lk];
            *(float4*)&As[r][lk] = va;
            const float4 vb = *(const float4*)&h[(size_t)(col0 + r) * DDIM + kb + lk];
            *(float4*)&Bs[r][lk] = vb;
        }
        __syncthreads();

#pragma unroll
        for (int kk = 0; kk < KC; kk += 4) {
            const int ka = kk + half * 2;                 // even -> b64 aligned
            const v2f a0 = *(const v2f*)&As[r0w + m16][ka];
            const v2f a1 = *(const v2f*)&As[r0w + 16 + m16][ka];
            v2f b[4];
#pragma unroll
            for (int t = 0; t < 4; ++t)
                b[t] = *(const v2f*)&Bs[c0w + t * 16 + m16][ka];
#pragma unroll
            for (int t = 0; t < 4; ++t) {
                acc[t] = __builtin_amdgcn_wmma_f32_16x16x4_f32(
                    false, a0, false, b[t], (short)0, acc[t], false, false);
                acc[4 + t] = __builtin_amdgcn_wmma_f32_16x16x4_f32(
                    false, a1, false, b[t], (short)0, acc[4 + t], false, false);
            }
        }
    }

    // Epilogue: apply rank-1 normalization and store.
    // C/D layout: VGPR v, lane -> M = v + half*8, N = m16
    float sr[16];
#pragma unroll
    for (int g = 0; g < 2; ++g)
#pragma unroll
        for (int v = 0; v < 8; ++v)
            sr[g * 8 + v] = scale[row0 + r0w + g * 16 + v + half * 8];

#pragma unroll
    for (int g = 0; g < 2; ++g) {
        const int crow = row0 + r0w + g * 16;
#pragma unroll
        for (int t = 0; t < 4; ++t) {
            const int cc = col0 + c0w + t * 16 + m16;
            const float sc = scale[cc];
#pragma unroll
            for (int v = 0; v < 8; ++v) {
                const int rr = crow + v + half * 8;
                simmat[(size_t)rr * KTOT + cc] = acc[g * 4 + t][v] * sr[g * 8 + v] * sc;
            }
        }
    }
}

// --------------------------------------------------------- per-row loss
__global__ __launch_bounds__(256) void rowloss_kernel(
        const float* __restrict__ simmat, float* __restrict__ out) {
    __shared__ float vals[KTOT];   // 8 KB
    __shared__ float red[256];
    const int i = blockIdx.x;
    const int imod = i & (BSZ - 1);
    const float* row = simmat + (size_t)i * KTOT;

    for (int j = threadIdx.x; j < KTOT; j += 256) vals[j] = row[j];
    __syncthreads();

    // max over negatives (j % BS != i % BS)
    float mx = -3.4e38f;
    for (int j = threadIdx.x; j < KTOT; j += 256)
        if ((j & (BSZ - 1)) != imod) mx = fmaxf(mx, vals[j]);
    red[threadIdx.x] = mx;
    __syncthreads();
    for (int s = 128; s > 0; s >>= 1) {
        if (threadIdx.x < s) red[threadIdx.x] = fmaxf(red[threadIdx.x], red[threadIdx.x + s]);
        __syncthreads();
    }
    mx = red[0];
    __syncthreads();

    // sum exp over negatives
    float se = 0.0f;
    for (int j = threadIdx.x; j < KTOT; j += 256)
        if ((j & (BSZ - 1)) != imod) se += __expf(vals[j] - mx);
    red[threadIdx.x] = se;
    __syncthreads();
    for (int s = 128; s > 0; s >>= 1) {
        if (threadIdx.x < s) red[threadIdx.x] += red[threadIdx.x + s];
        __syncthreads();
    }
    const float neg_lse = mx + __logf(red[0]);
    __syncthreads();

    // 15 positives: sum logaddexp(neg_lse, pos) - pos
    float c = 0.0f;
    if (threadIdx.x < N_T) {
        const int j = imod + threadIdx.x * BSZ;
        if (j != i) {
            const float p = vals[j];
            const float m = fmaxf(neg_lse, p);
            const float lae = m + __logf(__expf(neg_lse - m) + __expf(p - m));
            c = lae - p;
        }
    }
    red[threadIdx.x] = c;
    __syncthreads();
    for (int s = 128; s > 0; s >>= 1) {
        if (threadIdx.x < s) red[threadIdx.x] += red[threadIdx.x + s];
        __syncthreads();
    }
    if (threadIdx.x == 0)
        atomicAdd(out, red[0] * (1.0f / (float)(KTOT * (N_T - 1))));
}

// -------------------------------------------------- cyclic-shift MSE
__global__ __launch_bounds__(256) void mse_kernel(
        const float* __restrict__ pic, const float* __restrict__ dec,
        float* __restrict__ out) {
    __shared__ float red[256];
    const long long PER  = (long long)BSZ * PIX / 4;   // float4s per n-slice
    const long long TOT4 = (long long)N_T * PER;
    float s = 0.0f;
    for (long long e = (long long)blockIdx.x * 256 + threadIdx.x; e < TOT4;
         e += (long long)gridDim.x * 256) {
        const long long n   = e / PER;
        const long long rem = e - n * PER;
        const long long nd  = (n + 1) & (N_T - 1);
        const float4 p = ((const float4*)pic)[e];
        const float4 d = ((const float4*)dec)[nd * PER + rem];
        const float dx = p.x - d.x, dy = p.y - d.y, dz = p.z - d.z, dw = p.w - d.w;
        s += dx * dx + dy * dy + dz * dz + dw * dw;
    }
    red[threadIdx.x] = s;
    __syncthreads();
    for (int st = 128; st > 0; st >>= 1) {
        if (threadIdx.x < st) red[threadIdx.x] += red[threadIdx.x + st];
        __syncthreads();
    }
    if (threadIdx.x == 0)
        atomicAdd(out, red[0] * (1.0f / ((float)N_T * BSZ * PIX)));
}

// ---------------------------------------------------------------- launch
extern "C" void kernel_launch(void* const* d_in, const int* in_sizes, int n_in,
                              void* d_out, int out_size, void* d_ws, size_t ws_size,
                              hipStream_t stream) {
    const float* pic = (const float*)d_in[0];
    const float* dec = (const float*)d_in[1];
    const float* h   = (const float*)d_in[2];
    float* out = (float*)d_out;

    float* ws     = (float*)d_ws;
    float* scale  = ws;                 // 2048 floats
    float* simmat = ws + KTOT;          // 2048*2048 floats (16 MB)

    init_out_kernel<<<1, 32, 0, stream>>>(out);
    row_scale_kernel<<<KTOT, 256, 0, stream>>>(h, scale);
    gram_wmma_kernel<<<dim3(KTOT / TILE, KTOT / TILE), 256, 0, stream>>>(h, scale, simmat);
    rowloss_kernel<<<KTOT, 256, 0, stream>>>(simmat, out);
    mse_kernel<<<2048, 256, 0, stream>>>(pic, dec, out);
}